// DeepSeekV2Attention_39874476376642
// MI455X (gfx1250) — compile-verified
//
#include <hip/hip_runtime.h>
#include <math.h>

typedef __attribute__((ext_vector_type(2))) float v2f;
typedef __attribute__((ext_vector_type(8))) float v8f;
typedef int v4i_ __attribute__((__vector_size__(4 * sizeof(int))));

#define B_    128
#define MAXS  4096
#define HID_  5120
#define H_    128
#define DN    128
#define DR    64
#define DV    128
#define QCOLS (H_ * (DN + DR))   /* 24576 */
#define KVSZ  576
#define QN    (H_ * DN)          /* 16384: q_nope / attn inner dim */

#define AS1 __attribute__((address_space(1)))
#define AS3 __attribute__((address_space(3)))

#if defined(__AMDGCN__) && __has_builtin(__builtin_amdgcn_global_load_async_to_lds_b128)
#define USE_ASYNC 1
#else
#define USE_ASYNC 0
#endif

__device__ __forceinline__ v8f wmma4(v2f a, v2f b, v8f c) {
  // D = A(16x4 f32) * B(4x16 f32) + C(16x16 f32)
  return __builtin_amdgcn_wmma_f32_16x16x4_f32(
      /*neg_a=*/false, a, /*neg_b=*/false, b,
      /*c_mod=*/(short)0, c, /*reuse_a=*/false, /*reuse_b=*/false);
}

#if USE_ASYNC
__device__ __forceinline__ void cp_async16(const float* g, float* l) {
  // 16B global -> LDS, tracked by ASYNCcnt (emits global_load_async_to_lds_b128)
  __builtin_amdgcn_global_load_async_to_lds_b128(
      (AS1 v4i_*)(g), (AS3 v4i_*)(l), 0, 0);
}
__device__ __forceinline__ void wait_async0() {
#if __has_builtin(__builtin_amdgcn_s_wait_asynccnt)
  __builtin_amdgcn_s_wait_asynccnt(0);
#else
  asm volatile("s_wait_asynccnt 0" ::: "memory");
#endif
}
#endif

// ---------------------------------------------------------------------------
// Kernel 1: q_nope = h @ Wq[:, nope cols]  -> qws (B x (H*DN))
// grid (8, 256), block 128 (4 waves); each wave: one 16x16 tile, K-loop 5120.
// ---------------------------------------------------------------------------
__global__ __launch_bounds__(128)
void k_qgemm(const float* __restrict__ hs, const float* __restrict__ Wq,
             float* __restrict__ qws) {
  const int lane = threadIdx.x & 31;
  const int wv   = threadIdx.x >> 5;
  const int lm   = lane & 15, hi = lane >> 4;
  const int m0   = blockIdx.x * 16;               // batch tile
  const int t    = blockIdx.y * 4 + wv;           // N tile (0..1023)
  const int head = t >> 3;
  const int qc0  = head * (DN + DR) + (t & 7) * 16;  // Wq column base (skip rope)

  v8f acc = {0.f,0.f,0.f,0.f,0.f,0.f,0.f,0.f};
  const float* __restrict__ arow = hs + (size_t)(m0 + lm) * HID_;
  #pragma unroll 4
  for (int k = 0; k < HID_; k += 4) {
    v2f a, b;
    a.x = arow[k + 2*hi];
    a.y = arow[k + 2*hi + 1];
    b.x = Wq[(size_t)(k + 2*hi)     * QCOLS + qc0 + lm];
    b.y = Wq[(size_t)(k + 2*hi + 1) * QCOLS + qc0 + lm];
    acc = wmma4(a, b, acc);
  }
  #pragma unroll
  for (int r = 0; r < 8; ++r)
    qws[(size_t)(m0 + r + 8*hi) * QN + t*16 + lm] = acc[r];
}

// ---------------------------------------------------------------------------
// Kernel 2: kv = h @ Wkv[:, :256]; split into k_new (B x 128), v_new (B x 128)
// ---------------------------------------------------------------------------
__global__ __launch_bounds__(256)
void k_kv(const float* __restrict__ hs, const float* __restrict__ Wkv,
          float* __restrict__ knew, float* __restrict__ vnew) {
  __shared__ float hl[HID_];
  const int b = blockIdx.x;
  for (int i = threadIdx.x; i < HID_; i += 256)
    hl[i] = hs[(size_t)b * HID_ + i];
  __syncthreads();
  const int j = threadIdx.x;  // 0..255 -> first 256 cols of Wkv
  float s = 0.f;
  #pragma unroll 4
  for (int k = 0; k < HID_; ++k)
    s = fmaf(hl[k], Wkv[(size_t)k * KVSZ + j], s);
  if (j < DN) knew[b * DN + j]        = s;
  else        vnew[b * DV + (j - DN)] = s;
}

// ---------------------------------------------------------------------------
// Kernel 3: flash attention, one block per batch, 8 waves x 16 heads.
// K/V tiles double-buffered in LDS via global_load_async_to_lds_b128 (when
// available); scores and P*V via v_wmma_f32_16x16x4_f32; online softmax.
// ---------------------------------------------------------------------------
__global__ __launch_bounds__(256)
void k_attn(const float* __restrict__ qws, const float* __restrict__ kc,
            const float* __restrict__ vc,  const float* __restrict__ knew,
            const float* __restrict__ vnew, const int* __restrict__ seq_lens,
            const int* __restrict__ slot_map, float* __restrict__ aws) {
#if USE_ASYNC
  __shared__ float Kt[2][16 * 132];    // double-buffered K tile, pitch 132
  __shared__ float Vt[2][16 * 132];
#else
  __shared__ float Kt[1][16 * 132];
  __shared__ float Vt[1][16 * 132];
#endif
  __shared__ float Ps[8][16 * 17];     // per-wave P scratch, pitch 17

  const int b    = blockIdx.x;
  const int tid  = threadIdx.x;
  const int lane = tid & 31, wv = tid >> 5;
  const int lm   = lane & 15, hi = lane >> 4;
  const int slen = seq_lens[b];
  const int slot = slot_map[b];

  // Q fragments (16 heads x 128 d) in A-layout registers: M=lm, K=4c+v+2hi
  float qa[64];
  const float* __restrict__ qrow = qws + (size_t)b * QN + (wv * 16 + lm) * DN;
  #pragma unroll
  for (int c = 0; c < 32; ++c) {
    qa[2*c]     = qrow[4*c + 2*hi];
    qa[2*c + 1] = qrow[4*c + 2*hi + 1];
  }

  v8f o[8];
  #pragma unroll
  for (int n = 0; n < 8; ++n) o[n] = (v8f){0.f,0.f,0.f,0.f,0.f,0.f,0.f,0.f};
  float mr[8], lr[8];
  #pragma unroll
  for (int r = 0; r < 8; ++r) { mr[r] = -3.0e30f; lr[r] = 0.f; }

  const float scale  = 0.08838834764831845f;  // 128^-0.5
  const int   ntiles = (slen + 15) >> 4;
  const int   lrow   = tid >> 4;              // loader: row 0..15
  const int   lcol   = (tid & 15) * 8;        // loader: 8 floats per thread

#if USE_ASYNC
  // issue async copy of tile tI's 8 floats (K and V) into buffer `buf`
  auto issue_tile = [&](int tI, int buf) {
    const int srow = tI * 16 + lrow;
    const float* __restrict__ ksrc = (srow == slot)
        ? (knew + (size_t)b * DN) : (kc + ((size_t)b * MAXS + srow) * DN);
    const float* __restrict__ vsrc = (srow == slot)
        ? (vnew + (size_t)b * DV) : (vc + ((size_t)b * MAXS + srow) * DV);
    cp_async16(ksrc + lcol,     &Kt[buf][lrow * 132 + lcol]);
    cp_async16(ksrc + lcol + 4, &Kt[buf][lrow * 132 + lcol + 4]);
    cp_async16(vsrc + lcol,     &Vt[buf][lrow * 132 + lcol]);
    cp_async16(vsrc + lcol + 4, &Vt[buf][lrow * 132 + lcol + 4]);
  };
  issue_tile(0, 0);
#endif

  for (int tI = 0; tI < ntiles; ++tI) {
#if USE_ASYNC
    const int cur = tI & 1;
    wait_async0();                       // this wave's tile-tI copies landed
    __syncthreads();                     // all waves' copies visible
    if (tI + 1 < ntiles) issue_tile(tI + 1, 1 - cur);  // overlap next tile
#else
    const int cur = 0;
    {
      const int srow = tI * 16 + lrow;
      const float* __restrict__ ksrc = (srow == slot)
          ? (knew + (size_t)b * DN) : (kc + ((size_t)b * MAXS + srow) * DN);
      const float* __restrict__ vsrc = (srow == slot)
          ? (vnew + (size_t)b * DV) : (vc + ((size_t)b * MAXS + srow) * DV);
      #pragma unroll
      for (int e = 0; e < 8; ++e) {
        Kt[0][lrow * 132 + lcol + e] = ksrc[lcol + e];
        Vt[0][lrow * 132 + lcol + e] = vsrc[lcol + e];
      }
    }
    __syncthreads();
#endif

    // --- scores: S = Q (16h x 128d) * K^T (128d x 16s) ---
    const int s0 = tI * 16;
    v8f sc = {0.f,0.f,0.f,0.f,0.f,0.f,0.f,0.f};
    #pragma unroll
    for (int c = 0; c < 32; ++c) {
      v2f a, bb;
      a.x  = qa[2*c]; a.y = qa[2*c + 1];
      bb.x = Kt[cur][lm * 132 + 4*c + 2*hi];      // B: N=seq(lm), K=d
      bb.y = Kt[cur][lm * 132 + 4*c + 2*hi + 1];
      sc = wmma4(a, bb, sc);
    }

    const bool colValid = (s0 + lm) < slen;
    float sv[8], tmax[8], alpha[8], rs[8];
    #pragma unroll
    for (int r = 0; r < 8; ++r) sv[r] = colValid ? sc[r] * scale : -3.0e30f;

    // row max: 16 columns of a row live in one 16-lane half -> xor butterfly
    #pragma unroll
    for (int r = 0; r < 8; ++r) {
      float v = sv[r];
      v = fmaxf(v, __shfl_xor(v, 1, 32));
      v = fmaxf(v, __shfl_xor(v, 2, 32));
      v = fmaxf(v, __shfl_xor(v, 4, 32));
      v = fmaxf(v, __shfl_xor(v, 8, 32));
      tmax[r] = v;
    }
    #pragma unroll
    for (int r = 0; r < 8; ++r) {
      const float mnew = fmaxf(mr[r], tmax[r]);
      alpha[r] = __expf(mr[r] - mnew);
      mr[r] = mnew;
    }
    #pragma unroll
    for (int r = 0; r < 8; ++r) {
      float p = __expf(sv[r] - mr[r]);   // masked cols -> exp(-huge) = 0
      sv[r] = p;
      p += __shfl_xor(p, 1, 32);
      p += __shfl_xor(p, 2, 32);
      p += __shfl_xor(p, 4, 32);
      p += __shfl_xor(p, 8, 32);
      rs[r] = p;
    }
    #pragma unroll
    for (int r = 0; r < 8; ++r) lr[r] = lr[r] * alpha[r] + rs[r];
    #pragma unroll
    for (int n = 0; n < 8; ++n)
      #pragma unroll
      for (int r = 0; r < 8; ++r) o[n][r] *= alpha[r];

    // C-layout -> A-layout via per-wave LDS (wave-private, DScnt ordered)
    #pragma unroll
    for (int r = 0; r < 8; ++r) Ps[wv][(r + 8*hi) * 17 + lm] = sv[r];
    #pragma unroll
    for (int c = 0; c < 4; ++c) {
      v2f a;
      a.x = Ps[wv][lm * 17 + 4*c + 2*hi];
      a.y = Ps[wv][lm * 17 + 4*c + 2*hi + 1];
      #pragma unroll
      for (int n = 0; n < 8; ++n) {
        v2f bb;
        bb.x = Vt[cur][(4*c + 2*hi)     * 132 + n*16 + lm];  // B: K=seq, N=d
        bb.y = Vt[cur][(4*c + 2*hi + 1) * 132 + n*16 + lm];
        o[n] = wmma4(a, bb, o[n]);
      }
    }
    __syncthreads();
  }

  // epilogue: normalize and write attn (B x H x DV)
  #pragma unroll
  for (int n = 0; n < 8; ++n)
    #pragma unroll
    for (int r = 0; r < 8; ++r)
      aws[(size_t)b * QN + (wv*16 + r + 8*hi) * DV + n*16 + lm] =
          o[n][r] / lr[r];
}

// ---------------------------------------------------------------------------
// Kernel 4: out = attn (128 x 16384) @ Wo (16384 x 5120)
// grid (8, 80), block 128 (4 waves); one 16x16 tile per wave.
// ---------------------------------------------------------------------------
__global__ __launch_bounds__(128)
void k_ogemm(const float* __restrict__ aws, const float* __restrict__ Wo,
             float* __restrict__ out) {
  const int lane = threadIdx.x & 31;
  const int wv   = threadIdx.x >> 5;
  const int lm   = lane & 15, hi = lane >> 4;
  const int m0   = blockIdx.x * 16;
  const int n0   = (blockIdx.y * 4 + wv) * 16;

  v8f acc = {0.f,0.f,0.f,0.f,0.f,0.f,0.f,0.f};
  const float* __restrict__ arow = aws + (size_t)(m0 + lm) * QN;
  #pragma unroll 4
  for (int k = 0; k < QN; k += 4) {
    v2f a, b;
    a.x = arow[k + 2*hi];
    a.y = arow[k + 2*hi + 1];
    b.x = Wo[(size_t)(k + 2*hi)     * HID_ + n0 + lm];
    b.y = Wo[(size_t)(k + 2*hi + 1) * HID_ + n0 + lm];
    acc = wmma4(a, b, acc);
  }
  #pragma unroll
  for (int r = 0; r < 8; ++r)
    out[(size_t)(m0 + r + 8*hi) * HID_ + n0 + lm] = acc[r];
}

// ---------------------------------------------------------------------------
extern "C" void kernel_launch(void* const* d_in, const int* in_sizes, int n_in,
                              void* d_out, int out_size, void* d_ws, size_t ws_size,
                              hipStream_t stream) {
  const float* hs   = (const float*)d_in[0];  // (B,1,HID)
  const float* kc   = (const float*)d_in[1];  // (B,MAX_S,DN)
  const float* vc   = (const float*)d_in[2];  // (B,MAX_S,DV)
  const float* Wq   = (const float*)d_in[3];  // (HID, 24576)
  const float* Wkv  = (const float*)d_in[4];  // (HID, 576)
  const float* Wo   = (const float*)d_in[5];  // (16384, HID)
  // d_in[6] = positions (unused by the reference math)
  const int*   slot = (const int*)d_in[7];    // slot_mapping (B)
  const int*   slen = (const int*)d_in[8];    // seq_lens (B)
  float*       out  = (float*)d_out;

  float* ws   = (float*)d_ws;
  float* qws  = ws;                       // B*QN = 2,097,152 floats
  float* aws  = qws + (size_t)B_ * QN;    // B*QN
  float* knew = aws + (size_t)B_ * QN;    // B*DN
  float* vnew = knew + (size_t)B_ * DN;   // B*DV  (total ~16.1 MB)

  k_qgemm<<<dim3(8, 256), 128, 0, stream>>>(hs, Wq, qws);
  k_kv   <<<dim3(B_),     256, 0, stream>>>(hs, Wkv, knew, vnew);
  k_attn <<<dim3(B_),     256, 0, stream>>>(qws, kc, vc, knew, vnew, slen, slot, aws);
  k_ogemm<<<dim3(8, 80),  128, 0, stream>>>(aws, Wo, out);
}